// DeepPReLU_17841294148243
// MI455X (gfx1250) — compile-verified
//
#include <hip/hip_runtime.h>

typedef float v2f __attribute__((ext_vector_type(2)));
typedef float v8f __attribute__((ext_vector_type(8)));

#define NN   4096
#define NP1  4097
#define JBLK 64
#define NJB  65   // ceil(4097/64)
#define KC   16   // K-chunk staged in LDS per block iteration

// ---------------------------------------------------------------------------
// K1: fused 4-way fp32 WMMA GEMM (ReLU-split form) + step-2 column reduction.
//   new_lb = A_lb @ Bm + |A_lb| @ Bd        Bm = (W1lb+W1ub)/2
//   new_ub = A_ub @ Bm - |A_ub| @ Bd        Bd = (W1lb-W1ub)/2
// Block: 256 threads (8 waves), tile = 128 rows x 64 cols.
// B fragments are staged in LDS once per block (8 waves share them),
// stored as k-pairs (v2f) so compute reads are single ds_load_b64.
// ---------------------------------------------------------------------------
__global__ __launch_bounds__(256) void k1_gemm_fused(
    const float* __restrict__ W2lb, const float* __restrict__ W2ub,
    const float* __restrict__ W1lb, const float* __restrict__ W1ub,
    const float* __restrict__ lb0,  const float* __restrict__ ub0,
    float* __restrict__ p_lb, float* __restrict__ p_ub)
{
    __shared__ v2f sBm [(KC / 2) * JBLK];   // [k2][col]
    __shared__ v2f sBd [(KC / 2) * JBLK];
    __shared__ v2f sBdn[(KC / 2) * JBLK];

    const int tid  = threadIdx.x;
    const int wave = tid >> 5;
    const int lane = tid & 31;
    const int half = lane >> 4;     // K pair select / row+8 select
    const int l15  = lane & 15;

    const int iBase = blockIdx.y * 128;
    const int jBase = blockIdx.x * JBLK;
    const int rowW  = iBase + wave * 16;

    // ---- staging thread mapping: col = tid&63, k-pair = tid>>6 (+4 per pass)
    const int   scol = tid & 63;
    const int   sk2  = tid >> 6;                 // 0..3
    const int   gcol = jBase + scol;
    const int   colc = (gcol > NN) ? NN : gcol;  // clamp (in-bounds load)
    const float smsk = (gcol <= NN) ? 1.0f : 0.0f;
    const float* sW1lb = W1lb + colc;
    const float* sW1ub = W1ub + colc;

    v8f acc_lb[4], acc_ub[4];
#pragma unroll
    for (int t = 0; t < 4; ++t) {
        acc_lb[t] = (v8f){0.f,0.f,0.f,0.f,0.f,0.f,0.f,0.f};
        acc_ub[t] = (v8f){0.f,0.f,0.f,0.f,0.f,0.f,0.f,0.f};
    }

    // A fragment base: lane l15 -> row, half -> K offset {0,2}; v0/v1 = K, K+1.
    const int arow = rowW + l15;
    const float* aLbP = W2lb + (size_t)arow * NP1 + 2 * half;
    const float* aUbP = W2ub + (size_t)arow * NP1 + 2 * half;

    // ---- preload chunk-0 staging values into registers
    float rl0[2], rl1[2], ru0[2], ru1[2];
#pragma unroll
    for (int p = 0; p < 2; ++p) {
        const size_t gk = (size_t)(2 * (sk2 + 4 * p)) * NP1;
        rl0[p] = sW1lb[gk];       ru0[p] = sW1ub[gk];
        rl1[p] = sW1lb[gk + NP1]; ru1[p] = sW1ub[gk + NP1];
    }

    for (int k0 = 0; k0 < NN; k0 += KC) {
        // ---- store staged B chunk to LDS (Bm / Bd / -Bd), masked columns = 0
#pragma unroll
        for (int p = 0; p < 2; ++p) {
            const int k2 = sk2 + 4 * p;
            const float bl0 = rl0[p] * smsk, bl1 = rl1[p] * smsk;
            const float bu0 = ru0[p] * smsk, bu1 = ru1[p] * smsk;
            v2f bm, bd;
            bm[0] = 0.5f * (bl0 + bu0); bm[1] = 0.5f * (bl1 + bu1);
            bd[0] = 0.5f * (bl0 - bu0); bd[1] = 0.5f * (bl1 - bu1);
            sBm [k2 * JBLK + scol] = bm;
            sBd [k2 * JBLK + scol] = bd;
            sBdn[k2 * JBLK + scol] = (v2f){-bd[0], -bd[1]};
        }
        __syncthreads();

        // ---- prefetch next chunk's staging loads (overlap with WMMA below)
        if (k0 + KC < NN) {
#pragma unroll
            for (int p = 0; p < 2; ++p) {
                const size_t gk = (size_t)(k0 + KC + 2 * (sk2 + 4 * p)) * NP1;
                rl0[p] = sW1lb[gk];       ru0[p] = sW1ub[gk];
                rl1[p] = sW1lb[gk + NP1]; ru1[p] = sW1ub[gk + NP1];
            }
        }

        // ---- 4 k-steps of 4, 16 WMMAs each
#pragma unroll
        for (int s = 0; s < KC; s += 4) {
            v2f aL, aU, aLa, aUa;
            aL[0] = aLbP[k0 + s]; aL[1] = aLbP[k0 + s + 1];
            aU[0] = aUbP[k0 + s]; aU[1] = aUbP[k0 + s + 1];
            aLa[0] = __builtin_fabsf(aL[0]); aLa[1] = __builtin_fabsf(aL[1]);
            aUa[0] = __builtin_fabsf(aU[0]); aUa[1] = __builtin_fabsf(aU[1]);

            const int kidx = ((s >> 1) + half) * JBLK;
#pragma unroll
            for (int t = 0; t < 4; ++t) {
                const int idx = kidx + t * 16 + l15;
                const v2f bm  = sBm[idx];
                const v2f bd  = sBd[idx];
                const v2f bdn = sBdn[idx];
                acc_lb[t] = __builtin_amdgcn_wmma_f32_16x16x4_f32(
                    false, aL,  false, bm,  (short)0, acc_lb[t], false, false);
                acc_lb[t] = __builtin_amdgcn_wmma_f32_16x16x4_f32(
                    false, aLa, false, bd,  (short)0, acc_lb[t], false, false);
                acc_ub[t] = __builtin_amdgcn_wmma_f32_16x16x4_f32(
                    false, aU,  false, bm,  (short)0, acc_ub[t], false, false);
                acc_ub[t] = __builtin_amdgcn_wmma_f32_16x16x4_f32(
                    false, aUa, false, bdn, (short)0, acc_ub[t], false, false);
            }
        }
        __syncthreads();
    }

    // ---- fused step-2: weight columns by (m0, d0) and reduce over j ----
    float red_lb[8], red_ub[8];
#pragma unroll
    for (int r = 0; r < 8; ++r) { red_lb[r] = 0.f; red_ub[r] = 0.f; }

#pragma unroll
    for (int t = 0; t < 4; ++t) {
        const int col = jBase + t * 16 + l15;
        float m0 = 0.f, d0 = 0.f;
        if (col < NN)       { const float l = lb0[col], u = ub0[col];
                              m0 = 0.5f * (l + u); d0 = 0.5f * (l - u); }
        else if (col == NN) { m0 = 1.f; d0 = 0.f; }
#pragma unroll
        for (int r = 0; r < 8; ++r) {
            const float vl = acc_lb[t][r];
            const float vu = acc_ub[t][r];
            red_lb[r] += vl * m0 + __builtin_fabsf(vl) * d0;
            red_ub[r] += vu * m0 - __builtin_fabsf(vu) * d0;
        }
    }

    // Cross-lane reduce over the 16 columns held by each half-wave
    // (xor masks 1..8 stay within a 16-lane half). Deterministic order.
#pragma unroll
    for (int r = 0; r < 8; ++r) {
        float vl = red_lb[r], vu = red_ub[r];
        for (int m = 1; m < 16; m <<= 1) {
            vl += __shfl_xor(vl, m, 32);
            vu += __shfl_xor(vu, m, 32);
        }
        if (l15 == 0) {
            const int row = rowW + r + 8 * half;
            p_lb[(size_t)blockIdx.x * NN + row] = vl;
            p_ub[(size_t)blockIdx.x * NN + row] = vu;
        }
    }
}

// ---------------------------------------------------------------------------
// K2: reduce the 65 column-block partials per row (fixed order), add the
// layer-2 bias term, compute alpha / slope_ub / bias_ub per row.
// ---------------------------------------------------------------------------
__global__ __launch_bounds__(256) void k2_reduce(
    const float* __restrict__ p_lb, const float* __restrict__ p_ub,
    const float* __restrict__ W2lb, const float* __restrict__ W2ub,
    const float* __restrict__ alphas,
    float* __restrict__ alphaO, float* __restrict__ slopeO,
    float* __restrict__ biasO)
{
    const int i = blockIdx.x * 256 + threadIdx.x;
    if (i >= NN) return;

    // bias-row (k=4096) contributes W2*[i][4096] to column 4096, which
    // step-2 weights by m0=1, d0=0  ->  plain scalar add here.
    float slb = W2lb[(size_t)i * NP1 + NN];
    float sub = W2ub[(size_t)i * NP1 + NN];
    for (int jb = 0; jb < NJB; ++jb) {
        slb += p_lb[(size_t)jb * NN + i];
        sub += p_ub[(size_t)jb * NN + i];
    }

    float a = alphas[i];
    a = fminf(fmaxf(a, 0.f), 1.f);
    if (slb > 0.f)  a = 1.f;
    if (sub <= 0.f) a = 0.f;

    float s = sub / (sub - slb + 1.1920929e-07f);   // FLT_EPSILON
    if (slb > 0.f) s = 1.f;
    if (sub < 0.f) s = 0.f;

    float b = -slb * s;
    if (slb > 0.f || sub < 0.f) b = 0.f;

    alphaO[i] = a;
    slopeO[i] = s;
    biasO[i]  = b;
}

// ---------------------------------------------------------------------------
// K3: write the sparse (2, N, N+1) output: diagonals + bias column, rest 0.
// ---------------------------------------------------------------------------
__global__ __launch_bounds__(256) void k3_fill(
    const float* __restrict__ alphaO, const float* __restrict__ slopeO,
    const float* __restrict__ biasO, float* __restrict__ out)
{
    const int j     = blockIdx.x * 256 + threadIdx.x;
    const int i     = blockIdx.y;
    const int plane = blockIdx.z;
    if (j >= NP1) return;

    float v = 0.f;
    if (plane == 0) {
        if (j == i) v = alphaO[i];
    } else {
        if (j == i)       v = slopeO[i];
        else if (j == NN) v = biasO[i];
    }
    out[(size_t)plane * NN * NP1 + (size_t)i * NP1 + j] = v;
}

// ---------------------------------------------------------------------------
extern "C" void kernel_launch(void* const* d_in, const int* in_sizes, int n_in,
                              void* d_out, int out_size, void* d_ws, size_t ws_size,
                              hipStream_t stream)
{
    (void)in_sizes; (void)n_in; (void)out_size; (void)ws_size;

    const float* lb0    = (const float*)d_in[0];
    const float* ub0    = (const float*)d_in[1];
    const float* W1lb   = (const float*)d_in[2];
    const float* W1ub   = (const float*)d_in[3];
    const float* W2lb   = (const float*)d_in[4];
    const float* W2ub   = (const float*)d_in[5];
    const float* alphas = (const float*)d_in[6];
    float* out = (float*)d_out;

    float* ws     = (float*)d_ws;
    float* p_lb   = ws;                                  // NJB * N
    float* p_ub   = ws + (size_t)NJB * NN;               // NJB * N
    float* alphaO = ws + 2 * (size_t)NJB * NN;           // N
    float* slopeO = alphaO + NN;                         // N
    float* biasO  = slopeO + NN;                         // N

    dim3 g1(NJB, NN / 128);                // 65 x 32 blocks
    k1_gemm_fused<<<g1, 256, 0, stream>>>(W2lb, W2ub, W1lb, W1ub,
                                          lb0, ub0, p_lb, p_ub);

    k2_reduce<<<dim3(NN / 256), 256, 0, stream>>>(p_lb, p_ub, W2lb, W2ub,
                                                  alphas, alphaO, slopeO, biasO);

    k3_fill<<<dim3((NP1 + 255) / 256, NN, 2), 256, 0, stream>>>(
        alphaO, slopeO, biasO, out);
}